// BRIDGEDecoder_3719441678953
// MI455X (gfx1250) — compile-verified
//
#include <hip/hip_runtime.h>
#include <hip/hip_bf16.h>
#include <math.h>

// Sizes fixed by the reference
#define NB   16
#define NN   1024
#define FCH  256
#define NTOT (NB * NN)          // 16384 nodes
#define NE   524288             // edges
#define CONN (NB * NN * NN)     // 16777216 output elements
#define NEG_SLOPE 0.01f
#define EPSV 1e-10f

typedef __attribute__((ext_vector_type(2))) float v2f;
typedef __attribute__((ext_vector_type(8))) float v8f;

__device__ __forceinline__ float leaky(float v) { return v >= 0.f ? v : NEG_SLOPE * v; }

// ---------------------------------------------------------------- zero fill
__global__ void zero_f(float* __restrict__ p, int n) {
  int i = blockIdx.x * blockDim.x + threadIdx.x;
  if (i < n) p[i] = 0.f;
}

// ------------------------------------------------- param = relu(x @ W11^T + b11)
__global__ void param_kernel(const float* __restrict__ x, const float* __restrict__ W11,
                             const float* __restrict__ b11,
                             float* __restrict__ param, float* __restrict__ param_out) {
  int t = threadIdx.x;
  if (t < NB) {
    float acc = b11[0];
    for (int f = 0; f < FCH; ++f) acc = fmaf(x[t * FCH + f], W11[f], acc);
    float p = fmaxf(acc, 0.f);
    param[t] = p;
    param_out[t] = p;
  }
}

// --------------------------------------------- h = leaky(x @ W21^T + b21), [B*N]
__global__ __launch_bounds__(256) void h_kernel(const float* __restrict__ x,
                                                const float* __restrict__ W21,
                                                const float* __restrict__ b21,
                                                float* __restrict__ h) {
  __shared__ float xs[FCH];
  int i = blockIdx.x * 256 + threadIdx.x;  // 256 divides 1024 -> whole block same b
  int b = i >> 10;
  int n = i & 1023;
  xs[threadIdx.x] = x[b * FCH + threadIdx.x];
  __syncthreads();
  const float* wr = W21 + n * FCH;
  float acc = b21[n];
  for (int f = 0; f < FCH; ++f) acc = fmaf(xs[f], wr[f], acc);
  h[i] = leaky(acc);
}

// ---------------- edge pass 1: agg1[dst]+=w*h[src]; deg[dst]+=1; adj[b,s%N,d%N]+=w
__global__ void scatter1_kernel(const int* __restrict__ eidx, const float* __restrict__ ew,
                                const float* __restrict__ h,
                                float* __restrict__ agg1, float* __restrict__ deg,
                                float* __restrict__ adj) {
  int e = blockIdx.x * 256 + threadIdx.x;
  int s = eidx[e];
  int d = eidx[NE + e];
  float w = ew[e];
  atomicAdd(&agg1[d], w * h[s]);
  atomicAdd(&deg[d], 1.0f);
  size_t idx = ((size_t)(s >> 10) << 20) + (size_t)(s & 1023) * NN + (size_t)(d & 1023);
  atomicAdd(&adj[idx], w);
}

// ------------- h1 = leaky((agg1/deg)*Wrel1 + brel1 + h*Wroot1), [B*N, 8]
__global__ void h1_kernel(const float* __restrict__ agg1, const float* __restrict__ deg,
                          const float* __restrict__ h,
                          const float* __restrict__ Wrel1, const float* __restrict__ brel1,
                          const float* __restrict__ Wroot1, float* __restrict__ h1) {
  int i = blockIdx.x * 256 + threadIdx.x;
  float a = agg1[i] / fmaxf(deg[i], 1.0f);
  float hv = h[i];
#pragma unroll
  for (int c = 0; c < 8; ++c)
    h1[i * 8 + c] = leaky(fmaf(a, Wrel1[c], fmaf(hv, Wroot1[c], brel1[c])));
}

// ---------------- edge pass 2: agg2[dst, c] += w * h1[src, c]
__global__ void scatter2_kernel(const int* __restrict__ eidx, const float* __restrict__ ew,
                                const float* __restrict__ h1, float* __restrict__ agg2) {
  int e = blockIdx.x * 256 + threadIdx.x;
  int s = eidx[e];
  int d = eidx[NE + e];
  float w = ew[e];
#pragma unroll
  for (int c = 0; c < 8; ++c)
    atomicAdd(&agg2[d * 8 + c], w * h1[s * 8 + c]);
}

// ------------- out = leaky((agg2/deg)@Wrel2^T + brel2 + h1@Wroot2^T), [B*N, 16]
__global__ void out16_kernel(const float* __restrict__ agg2, const float* __restrict__ deg,
                             const float* __restrict__ h1,
                             const float* __restrict__ Wrel2, const float* __restrict__ brel2,
                             const float* __restrict__ Wroot2, float* __restrict__ o16) {
  int i = blockIdx.x * 256 + threadIdx.x;
  float invd = 1.0f / fmaxf(deg[i], 1.0f);
  float a[8], hv[8];
#pragma unroll
  for (int c = 0; c < 8; ++c) { a[c] = agg2[i * 8 + c] * invd; hv[c] = h1[i * 8 + c]; }
#pragma unroll
  for (int j = 0; j < 16; ++j) {
    float acc = brel2[j];
#pragma unroll
    for (int c = 0; c < 8; ++c)
      acc = fmaf(a[c], Wrel2[j * 8 + c], fmaf(hv[c], Wroot2[j * 8 + c], acc));
    o16[i * 16 + j] = leaky(acc);
  }
}

// ------------------------------------------------------------------ final kernel:
// per 16x16 tile (one wave32 each): k = A x B via 4 chained V_WMMA_F32_16X16X4_F32
// (A = out rows, B = out-tile transposed), then fused epilogue
//   conn = sigmoid(k + log(adj+eps)*param) * (S != 0)
// adj was accumulated in-place in `conn` (d_out); each element is RMW'd by one lane.
__global__ __launch_bounds__(256) void conn_wmma_kernel(const float* __restrict__ o16,
                                                        const float* __restrict__ param,
                                                        float* __restrict__ conn) {
  int wave = (int)((blockIdx.x * blockDim.x + threadIdx.x) >> 5);
  int lane = threadIdx.x & 31;
  int b  = wave >> 12;        // 4096 tiles per batch
  int r  = wave & 4095;
  int ti = r >> 6;
  int tj = r & 63;

  const float* ob = o16 + (size_t)b * NN * 16;
  int row0 = ti * 16, col0 = tj * 16;
  int half = lane >> 4, l16 = lane & 15;

  // A 16x4 f32 layout: lanes 0-15 {VGPR0=K0, VGPR1=K1}, lanes 16-31 {K2, K3}.
  // B 4x16 mirrors the K split; B[k][n] = out[col0+n][d] -> same formula, col0 base.
  const float* arow = ob + (size_t)(row0 + l16) * 16;
  const float* brow = ob + (size_t)(col0 + l16) * 16;

  v8f acc = {0.f, 0.f, 0.f, 0.f, 0.f, 0.f, 0.f, 0.f};
#pragma unroll
  for (int kk = 0; kk < 4; ++kk) {
    int d = kk * 4 + 2 * half;
    v2f av; av.x = arow[d]; av.y = arow[d + 1];
    v2f bv; bv.x = brow[d]; bv.y = brow[d + 1];
    acc = __builtin_amdgcn_wmma_f32_16x16x4_f32(
        /*neg_a=*/false, av, /*neg_b=*/false, bv,
        /*c_mod=*/(short)0, acc, /*reuse_a=*/false, /*reuse_b=*/false);
  }

  float p = param[b];
#pragma unroll
  for (int rr = 0; rr < 8; ++rr) {               // C/D: M = rr + 8*half, N = l16
    int m = row0 + rr + 8 * half;
    int nn = col0 + l16;
    size_t idx = ((size_t)b << 20) + (size_t)m * NN + nn;
    float S = conn[idx] + EPSV;                  // adj (+eps) accumulated here
    float z = acc[rr] + logf(S) * p;
    float v = 1.0f / (1.0f + expf(-z));
    conn[idx] = (S != 0.0f) ? v : 0.0f;
  }
}

extern "C" void kernel_launch(void* const* d_in, const int* in_sizes, int n_in,
                              void* d_out, int out_size, void* d_ws, size_t ws_size,
                              hipStream_t stream) {
  (void)in_sizes; (void)n_in; (void)out_size; (void)ws_size;

  const float* x      = (const float*)d_in[0];
  const int*   eidx   = (const int*)d_in[1];   // [2, E]: src then dst
  const float* ew     = (const float*)d_in[2];
  /* d_in[3] = batch (unused; g == src >> 10 by construction) */
  const float* W11    = (const float*)d_in[4];
  const float* b11    = (const float*)d_in[5];
  const float* W21    = (const float*)d_in[6];
  const float* b21    = (const float*)d_in[7];
  const float* Wrel1  = (const float*)d_in[8];
  const float* brel1  = (const float*)d_in[9];
  const float* Wroot1 = (const float*)d_in[10];
  const float* Wrel2  = (const float*)d_in[11];
  const float* brel2  = (const float*)d_in[12];
  const float* Wroot2 = (const float*)d_in[13];

  float* conn      = (float*)d_out;       // [B*N*N], first used as adj accumulator
  float* param_out = conn + CONN;         // [B]

  // workspace layout (floats): ~2.3 MB total
  float* ws    = (float*)d_ws;
  float* param = ws;                      // 16
  float* h     = param + NB;              // 16384
  float* deg   = h + NTOT;                // 16384   } zeroed together
  float* agg1  = deg + NTOT;              // 16384   }
  float* h1    = agg1 + NTOT;             // 131072
  float* agg2  = h1 + NTOT * 8;           // 131072
  float* o16   = agg2 + NTOT * 8;         // 262144

  // zero accumulators every call (deterministic; harness does not re-poison)
  zero_f<<<(CONN + 255) / 256, 256, 0, stream>>>(conn, CONN);
  zero_f<<<(2 * NTOT + 255) / 256, 256, 0, stream>>>(deg, 2 * NTOT);      // deg + agg1
  zero_f<<<(NTOT * 8 + 255) / 256, 256, 0, stream>>>(agg2, NTOT * 8);

  param_kernel<<<1, 32, 0, stream>>>(x, W11, b11, param, param_out);
  h_kernel<<<NTOT / 256, 256, 0, stream>>>(x, W21, b21, h);
  scatter1_kernel<<<NE / 256, 256, 0, stream>>>(eidx, ew, h, agg1, deg, conn);
  h1_kernel<<<NTOT / 256, 256, 0, stream>>>(agg1, deg, h, Wrel1, brel1, Wroot1, h1);
  scatter2_kernel<<<NE / 256, 256, 0, stream>>>(eidx, ew, h1, agg2);
  out16_kernel<<<NTOT / 256, 256, 0, stream>>>(agg2, deg, h1, Wrel2, brel2, Wroot2, o16);

  // 65536 tiles, one wave32 each, 8 waves/block -> 8192 blocks
  conn_wmma_kernel<<<(NB * 64 * 64) / 8, 256, 0, stream>>>(o16, param, conn);
}